// Model_25134148616671
// MI455X (gfx1250) — compile-verified
//
#include <hip/hip_runtime.h>
#include <math.h>

// ---------------------------------------------------------------------------
// Shapes
// ---------------------------------------------------------------------------
#define B_      16
#define L_      336
#define P_      96
#define M_      64
#define E_      64
#define NPATCH_ 41
#define R_      1024          // B*M rows for all MLP GEMMs
#define NPR_    6896          // B*(L+P-1) rows for prior
#define TM1_    431           // (L+P)-1 per batch

typedef __attribute__((ext_vector_type(2))) float v2f;
typedef __attribute__((ext_vector_type(8))) float v8f;

#define WPB 8   // waves per block (256 threads)

__device__ __forceinline__ float gelu_exact(float x) {
  return 0.5f * x * (1.0f + erff(x * 0.70710678118654752f));
}

// ---------------------------------------------------------------------------
// Generic fp32 WMMA GEMM:  C[R,N] = act(A[R,K] @ W[K,N] + bias)
// One wave computes a 16x(CT*16) strip of C; the A fragment is loaded once
// per K-step and reused across CT column tiles.  CT is a template constant
// chosen so that (N/16) % CT == 0 -> no conditionals in the hot loop, EXEC
// stays all-1s around every WMMA.
// ---------------------------------------------------------------------------
template <int CT>
__global__ __launch_bounds__(256) void gemm_wmma_f32(
    const float* __restrict__ A, const float* __restrict__ W,
    const float* __restrict__ bias, float* __restrict__ C,
    int R, int K, int N, int act)
{
  const int wid  = threadIdx.x >> 5;
  const int lane = threadIdx.x & 31;
  const int ctiles  = N >> 4;
  const int cgroups = ctiles / CT;             // exact by construction
  const int total   = (R >> 4) * cgroups;
  const int t = blockIdx.x * WPB + wid;
  if (t >= total) return;                      // whole wave exits together
  const int cg  = t % cgroups;
  const int r0  = (t / cgroups) << 4;
  const int c0  = cg * (CT * 16);
  const int l16  = lane & 15;
  const int koff = (lane >> 4) << 1;           // 0 or 2 (K split across lane halves)
  const float* arow = A + (size_t)(r0 + l16) * K + koff;
  const float* wcol = W + c0 + l16;
  v8f acc[CT];
#pragma unroll
  for (int ct = 0; ct < CT; ++ct) acc[ct] = (v8f){};
  for (int k = 0; k < K; k += 4) {
    v2f a;
    a.x = arow[k];
    a.y = arow[k + 1];
    const float* wk0 = wcol + (size_t)(k + koff) * N;
    const float* wk1 = wcol + (size_t)(k + koff + 1) * N;
#pragma unroll
    for (int ct = 0; ct < CT; ++ct) {
      v2f b;
      b.x = wk0[ct << 4];
      b.y = wk1[ct << 4];
      acc[ct] = __builtin_amdgcn_wmma_f32_16x16x4_f32(false, a, false, b,
                                                      (short)0, acc[ct], false, false);
    }
  }
  const int rbase = r0 + ((lane >> 4) << 3);   // lanes 16-31 hold rows M=v+8
#pragma unroll
  for (int ct = 0; ct < CT; ++ct) {
    const int cc = c0 + (ct << 4) + l16;
    const float bv = bias ? bias[cc] : 0.0f;
#pragma unroll
    for (int v = 0; v < 8; ++v) {
      float x = acc[ct][v] + bv;
      if (act) x = gelu_exact(x);
      C[(size_t)(rbase + v) * N + cc] = x;
    }
  }
}

// ---------------------------------------------------------------------------
// Per-row LayerNorm (in place), one wave per row, wave32 shuffles.
// ---------------------------------------------------------------------------
__global__ __launch_bounds__(256) void ln_rows(
    float* __restrict__ X, const float* __restrict__ g,
    const float* __restrict__ beta, int R, int N)
{
  const int wid  = threadIdx.x >> 5;
  const int lane = threadIdx.x & 31;
  const int row  = blockIdx.x * WPB + wid;
  if (row >= R) return;
  float* x = X + (size_t)row * N;
  float s = 0.f;
  for (int i = lane; i < N; i += 32) s += x[i];
  for (int o = 16; o; o >>= 1) s += __shfl_xor(s, o, 32);
  const float mu = s / (float)N;
  float v = 0.f;
  for (int i = lane; i < N; i += 32) { float d = x[i] - mu; v += d * d; }
  for (int o = 16; o; o >>= 1) v += __shfl_xor(v, o, 32);
  const float rs = rsqrtf(v / (float)N + 1e-5f);
  for (int i = lane; i < N; i += 32)
    x[i] = (x[i] - mu) * rs * g[i] + beta[i];
}

// ---------------------------------------------------------------------------
// RevIN: stats over L per (b,m), write normalized transposed XC[(b*64+m), L]
// ---------------------------------------------------------------------------
__global__ __launch_bounds__(64) void revin_kernel(
    const float* __restrict__ x, const float* __restrict__ rw,
    const float* __restrict__ rb, float* __restrict__ mean,
    float* __restrict__ stdev, float* __restrict__ XC)
{
  const int b = blockIdx.x, m = threadIdx.x;
  const float* xb = x + (size_t)b * L_ * M_ + m;
  float s = 0.f, s2 = 0.f;
  for (int l = 0; l < L_; ++l) { float v = xb[(size_t)l * M_]; s += v; s2 += v * v; }
  const float mu  = s / (float)L_;
  const float var = fmaxf(s2 / (float)L_ - mu * mu, 0.f);
  const float sd  = sqrtf(var + 1e-5f);
  const int r = b * M_ + m;
  mean[r] = mu; stdev[r] = sd;
  const float w = rw[m], bb = rb[m], inv = 1.0f / sd;
  float* o = XC + (size_t)r * L_;
  for (int l = 0; l < L_; ++l)
    o[l] = (xb[(size_t)l * M_] - mu) * inv * w + bb;
}

// patches: PM[(r*41+p)*16+j] = XC[r*336 + p*8 + j]
__global__ void make_patches(const float* __restrict__ XC, float* __restrict__ PM) {
  const int idx = blockIdx.x * blockDim.x + threadIdx.x;
  if (idx >= R_ * NPATCH_ * 16) return;
  const int j = idx & 15;
  const int q = idx >> 4;
  const int p = q % NPATCH_;
  const int r = q / NPATCH_;
  PM[idx] = XC[(size_t)r * L_ + p * 8 + j];
}

// reparam: Z[r,l] = X[r,l] + X[r,half+l] * eps[b,l,m]   (r = b*64+m)
__global__ void reparam_kernel(const float* __restrict__ X,
                               const float* __restrict__ eps,
                               float* __restrict__ Z, int half)
{
  const int idx = blockIdx.x * blockDim.x + threadIdx.x;
  if (idx >= R_ * half) return;
  const int l = idx % half;
  const int r = idx / half;
  const int b = r >> 6, m = r & 63;
  Z[idx] = X[(size_t)r * (2 * half) + l]
         + X[(size_t)r * (2 * half) + half + l] * eps[((size_t)b * half + l) * M_ + m];
}

// denorm: out[b,l,m] = (Y[r,l]-rb[m])/(rw[m]+1e-10)*stdev[r]+mean[r]
__global__ void denorm_kernel(const float* __restrict__ Y,
                              const float* __restrict__ rw, const float* __restrict__ rb,
                              const float* __restrict__ mean, const float* __restrict__ stdev,
                              float* __restrict__ out, int len)
{
  const int idx = blockIdx.x * blockDim.x + threadIdx.x;
  if (idx >= B_ * len * M_) return;
  const int m = idx & 63;
  const int t = idx >> 6;
  const int l = t % len;
  const int b = t / len;
  const int r = b * M_ + m;
  out[idx] = (Y[(size_t)r * len + l] - rb[m]) / (rw[m] + 1e-10f) * stdev[r] + mean[r];
}

// ---------------------------------------------------------------------------
// Prior GEMM (WMMA): PRE[n, i*4+h] = sum_f xx[n,f] * g_W1[i,h,f]
// xx gathered from ZR_s / ZP_s (src_z layout).  6896 x 64 x 256.
// One wave computes a 16x64 strip (4 accumulators, exactly 4 column groups).
// ---------------------------------------------------------------------------
__global__ __launch_bounds__(256) void prior_gemm_wmma(
    const float* __restrict__ ZRs, const float* __restrict__ ZPs,
    const float* __restrict__ gW1, float* __restrict__ PRE)
{
  const int wid  = threadIdx.x >> 5;
  const int lane = threadIdx.x & 31;
  const int t = blockIdx.x * WPB + wid;
  if (t >= TM1_ * 4) return;           // 431 row tiles x 4 column groups
  const int cg = t & 3;
  const int r0 = (t >> 2) << 4;
  const int c0 = cg << 6;
  const int l16  = lane & 15;
  const int koff = (lane >> 4) << 1;
  const int n  = r0 + l16;
  const int b  = n / TM1_;
  const int tt = n % TM1_;
  const float* zr = ZRs + (size_t)b * M_ * L_ + tt;
  const float* zp = ZPs + (size_t)b * M_ * P_ + (tt - L_);
  v8f acc[4];
#pragma unroll
  for (int ct = 0; ct < 4; ++ct) acc[ct] = (v8f){};
  for (int k = 0; k < M_; k += 4) {
    const int f0 = k + koff;
    v2f a;
    if (tt < L_) {
      a.x = zr[(size_t)f0 * L_];
      a.y = zr[(size_t)(f0 + 1) * L_];
    } else {
      a.x = zp[(size_t)f0 * P_];
      a.y = zp[(size_t)(f0 + 1) * P_];
    }
#pragma unroll
    for (int ct = 0; ct < 4; ++ct) {
      v2f bb;
      bb.x = gW1[(size_t)(c0 + (ct << 4) + l16) * 65 + f0];
      bb.y = gW1[(size_t)(c0 + (ct << 4) + l16) * 65 + f0 + 1];
      acc[ct] = __builtin_amdgcn_wmma_f32_16x16x4_f32(false, a, false, bb,
                                                      (short)0, acc[ct], false, false);
    }
  }
  const int rbase = r0 + ((lane >> 4) << 3);
#pragma unroll
  for (int ct = 0; ct < 4; ++ct)
#pragma unroll
    for (int v = 0; v < 8; ++v)
      PRE[(size_t)(rbase + v) * 256 + c0 + (ct << 4) + l16] = acc[ct][v];
}

__global__ void init_lad(float* __restrict__ lad) {
  if (threadIdx.x < B_) lad[threadIdx.x] = 0.0f;
}

// Per (n,i): residuals, jac store, logabsdet accumulation.
__global__ void prior_point(
    const float* __restrict__ ZRs, const float* __restrict__ ZPs,
    const float* __restrict__ PRE, const float* __restrict__ gW1,
    const float* __restrict__ gb1, const float* __restrict__ ga,
    const float* __restrict__ gW2, const float* __restrict__ gb2,
    float* __restrict__ resid, float* __restrict__ lad, float* __restrict__ JAC)
{
  const int idx = blockIdx.x * blockDim.x + threadIdx.x;
  if (idx >= NPR_ * M_) return;
  const int i  = idx & 63;
  const int n  = idx >> 6;
  const int b  = n / TM1_;
  const int t1 = (n % TM1_) + 1;
  const float yy = (t1 < L_) ? ZRs[((size_t)(b * M_ + i)) * L_ + t1]
                             : ZPs[((size_t)(b * M_ + i)) * P_ + (t1 - L_)];
  float res = gb2[i];
  float jl  = 0.f;
  const float av = ga[i];
  float jac[4];
  for (int h = 0; h < 4; ++h) {
    const int ih = i * 4 + h;
    const float w1y = gW1[(size_t)ih * 65 + 64];
    const float pre = PRE[(size_t)n * 256 + ih] + yy * w1y + gb1[ih];
    const float s   = (pre > 0.f) ? 1.f : av;
    const float w2  = gW2[ih];
    res += w2 * pre * s;
    const float j = w2 * s;
    jac[h] = j;
    jl += j * w1y;
  }
  resid[idx] = res;
  for (int h = 0; h < 4; ++h) JAC[(size_t)idx * 4 + h] = jac[h];
  atomicAdd(&lad[b], logf(fabsf(jl)));
}

// hist_jac[n,i,f] = sum_h JAC[n,i,h] * g_W1[i,h,f]   (bandwidth-bound, 113 MB out)
__global__ void hist_jac_kernel(const float* __restrict__ JAC,
                                const float* __restrict__ gW1,
                                float* __restrict__ out)
{
  const size_t idx = (size_t)blockIdx.x * blockDim.x + threadIdx.x;
  if (idx >= (size_t)NPR_ * M_ * M_) return;
  const int f = (int)(idx & 63);
  const size_t q = idx >> 6;          // (n,i)
  const int i = (int)(q & 63);
  const float* j4 = JAC + q * 4;
  const float* w  = gW1 + (size_t)i * 4 * 65 + f;
  out[idx] = j4[0] * w[0] + j4[1] * w[65] + j4[2] * w[130] + j4[3] * w[195];
}

// ---------------------------------------------------------------------------
// Host side
// ---------------------------------------------------------------------------
static inline int cdiv(long long a, int b) { return (int)((a + b - 1) / b); }

static void launch_gemm(const float* A, const float* W, const float* bias, float* C,
                        int R, int K, int N, int act, hipStream_t s) {
  const int ctiles = N / 16;
  int ct = (ctiles % 4 == 0) ? 4 : (ctiles % 3 == 0) ? 3 : (ctiles % 2 == 0) ? 2 : 1;
  const int tiles = (R / 16) * (ctiles / ct);
  const int blocks = cdiv(tiles, WPB);
  switch (ct) {
    case 4: gemm_wmma_f32<4><<<blocks, 256, 0, s>>>(A, W, bias, C, R, K, N, act); break;
    case 3: gemm_wmma_f32<3><<<blocks, 256, 0, s>>>(A, W, bias, C, R, K, N, act); break;
    case 2: gemm_wmma_f32<2><<<blocks, 256, 0, s>>>(A, W, bias, C, R, K, N, act); break;
    default: gemm_wmma_f32<1><<<blocks, 256, 0, s>>>(A, W, bias, C, R, K, N, act); break;
  }
}
static void launch_ln(float* X, const float* g, const float* b, int R, int N, hipStream_t s) {
  ln_rows<<<cdiv(R, WPB), 256, 0, s>>>(X, g, b, R, N);
}

// workspace offsets (floats)
#define O_MEAN   0
#define O_STD    1024
#define O_XC     2048          // 344064
#define O_PM     346112       // 671744
#define O_E      1017856      // 2686976
#define O_EMB    3704832      // 344064
#define O_T1     4048896      // 688128
#define O_ZED    4737024      // 688128
#define O_ZRS    5425152      // 344064
#define O_ZRT    5769216      // 344064
#define O_HP1    6113280      // 196608
#define O_HP     6309888      // 196608
#define O_ZPS    6506496      // 98304
#define O_ZPT    6604800      // 98304
#define O_HR1    6703104      // 688128
#define O_REC    7391232      // 344064
#define O_HD1    7735296      // 196608
#define O_PRED   7931904      // 98304
#define O_PRE    8030208      // 1765376
#define O_JAC    9795584      // 1765376  -> total 11560960 floats (~46.2 MB)

// output offsets (floats)
#define OO_SREC  0            // 344064
#define OO_SPRED 344064       // 98304
#define OO_TREC  442368       // 344064
#define OO_TPRED 786432       // 98304
#define OO_RES   884736       // 441344
#define OO_LAD   1326080      // 16
#define OO_HIST  1326096      // 28246016

extern "C" void kernel_launch(void* const* d_in, const int* in_sizes, int n_in,
                              void* d_out, int out_size, void* d_ws, size_t ws_size,
                              hipStream_t stream) {
  (void)in_sizes; (void)n_in; (void)out_size; (void)ws_size;
  const float* in[42];
  for (int i = 0; i < 42; ++i) in[i] = (const float*)d_in[i];

  const float* rev_w = in[7];
  const float* rev_b = in[8];
  float* ws  = (float*)d_ws;
  float* out = (float*)d_out;

  for (int s = 0; s < 2; ++s) {
    const float* x        = s ? in[1] : in[0];
    const float* eps_rec  = s ? in[5] : in[3];
    const float* eps_pred = s ? in[6] : in[4];
    float* ZR = ws + (s ? O_ZRT : O_ZRS);
    float* ZP = ws + (s ? O_ZPT : O_ZPS);
    float* out_rec  = out + (s ? OO_TREC  : OO_SREC);
    float* out_pred = out + (s ? OO_TPRED : OO_SPRED);

    // RevIN + transpose
    revin_kernel<<<B_, M_, 0, stream>>>(x, rev_w, rev_b,
        ws + O_MEAN, ws + O_STD, ws + O_XC);
    // patch embed
    make_patches<<<cdiv(R_ * NPATCH_ * 16, 256), 256, 0, stream>>>(ws + O_XC, ws + O_PM);
    launch_gemm(ws + O_PM, in[9],  in[10], ws + O_E,   R_ * NPATCH_, 16,   E_,  0, stream);
    launch_gemm(ws + O_E,  in[11], in[12], ws + O_EMB, R_, NPATCH_ * E_, L_, 0, stream);
    // encoder MLP2
    launch_gemm(ws + O_EMB, in[13], in[14], ws + O_T1, R_, L_,      2 * L_, 1, stream);
    launch_ln(ws + O_T1, in[15], in[16], R_, 2 * L_, stream);
    launch_gemm(ws + O_T1, in[17], in[18], ws + O_ZED, R_, 2 * L_, 2 * L_, 0, stream);
    reparam_kernel<<<cdiv(R_ * L_, 256), 256, 0, stream>>>(ws + O_ZED, eps_rec, ZR, L_);
    // pre MLP2 -> z_pred
    launch_gemm(ZR, in[19], in[20], ws + O_HP1, R_, L_,     2 * P_, 1, stream);
    launch_ln(ws + O_HP1, in[21], in[22], R_, 2 * P_, stream);
    launch_gemm(ws + O_HP1, in[23], in[24], ws + O_HP, R_, 2 * P_, 2 * P_, 0, stream);
    reparam_kernel<<<cdiv(R_ * P_, 256), 256, 0, stream>>>(ws + O_HP, eps_pred, ZP, P_);
    // rec decoder
    launch_gemm(ZR, in[25], in[26], ws + O_HR1, R_, L_,     2 * L_, 1, stream);
    launch_ln(ws + O_HR1, in[27], in[28], R_, 2 * L_, stream);
    launch_gemm(ws + O_HR1, in[29], in[30], ws + O_REC, R_, 2 * L_, L_, 0, stream);
    denorm_kernel<<<cdiv(B_ * L_ * M_, 256), 256, 0, stream>>>(ws + O_REC,
        rev_w, rev_b, ws + O_MEAN, ws + O_STD, out_rec, L_);
    // pred decoder
    launch_gemm(ZP, in[31], in[32], ws + O_HD1, R_, P_,     2 * P_, 1, stream);
    launch_ln(ws + O_HD1, in[33], in[34], R_, 2 * P_, stream);
    launch_gemm(ws + O_HD1, in[35], in[36], ws + O_PRED, R_, 2 * P_, P_, 0, stream);
    denorm_kernel<<<cdiv(B_ * P_ * M_, 256), 256, 0, stream>>>(ws + O_PRED,
        rev_w, rev_b, ws + O_MEAN, ws + O_STD, out_pred, P_);
  }

  // ---- prior (src stream only) ----
  init_lad<<<1, 32, 0, stream>>>(out + OO_LAD);
  prior_gemm_wmma<<<cdiv(TM1_ * 4, WPB), 256, 0, stream>>>(
      ws + O_ZRS, ws + O_ZPS, in[37], ws + O_PRE);
  prior_point<<<cdiv(NPR_ * M_, 256), 256, 0, stream>>>(
      ws + O_ZRS, ws + O_ZPS, ws + O_PRE, in[37], in[38], in[39], in[40], in[41],
      out + OO_RES, out + OO_LAD, ws + O_JAC);
  hist_jac_kernel<<<cdiv((long long)NPR_ * M_ * M_, 256), 256, 0, stream>>>(
      ws + O_JAC, in[37], out + OO_HIST);
}